// Minibatch_Discrimination_42829413876113
// MI455X (gfx1250) — compile-verified
//
#include <hip/hip_runtime.h>
#include <hip/hip_bf16.h>

// Problem constants (from reference)
#define B_SZ  64
#define K_DIM 8192            // N_FEAT * H * W
#define N_DIM 20480           // HID * H * W
#define HW    1024
#define HID   20

typedef __attribute__((ext_vector_type(2))) float v2f;
typedef __attribute__((ext_vector_type(8))) float v8f;

// ---------------------------------------------------------------------------
// Phase 1: h = x(64x8192) @ T(8192x20480), f32, V_WMMA_F32_16X16X4_F32.
// One wave per 32-column strip of h; each wave computes 4 M-tiles x 2 N-tiles
// (the full M=64), so B operands are reused by 4 WMMAs and A by 2.
// T is streamed non-temporally (read exactly once, 671 MB -> HBM-bound floor);
// x (2 MB) stays resident in L2 across all 640 waves.
//
// VGPR layouts per ISA 7.12.2:
//   A 16x4 f32: vgpr0 = K even (k+0 lanes 0-15, k+2 lanes 16-31), vgpr1 = K odd
//     -> one float2 load at (row, k + 2*half) fills both VGPRs.
//   B 4x16 f32: lanes 0-15 hold K=0,1; lanes 16-31 hold K=2,3 (N = lane&15).
//   C/D 16x16 f32: vgpr r holds M = r + 8*half, N = lane&15.
// ---------------------------------------------------------------------------
__global__ __launch_bounds__(256) void gemm_wmma_f32(const float* __restrict__ A,
                                                     const float* __restrict__ T,
                                                     float* __restrict__ H) {
  const int lane = threadIdx.x & 31;
  const int wave = (blockIdx.x * blockDim.x + threadIdx.x) >> 5;
  const int n0   = wave * 32;      // this wave's 32-column strip
  const int half = lane >> 4;      // 0: lanes 0-15, 1: lanes 16-31
  const int l15  = lane & 15;

  v8f acc[4][2] = {};              // 4 M-tiles x 2 N-tiles, zero-initialized C

  #pragma unroll 2
  for (int k = 0; k < K_DIM; k += 4) {
    // A operands: float2 per lane -> (vgpr0, vgpr1) = (K even, K odd)
    v2f a[4];
    #pragma unroll
    for (int m = 0; m < 4; ++m) {
      const int row = m * 16 + l15;
      a[m] = *(const v2f*)&A[(size_t)row * K_DIM + k + 2 * half];
    }
    // B operands: strided rows of T (non-temporal stream)
    v2f b[2];
    #pragma unroll
    for (int n = 0; n < 2; ++n) {
      const int col = n0 + n * 16 + l15;
      const int kr  = k + 2 * half;
      b[n].x = __builtin_nontemporal_load(&T[(size_t)kr * N_DIM + col]);
      b[n].y = __builtin_nontemporal_load(&T[(size_t)(kr + 1) * N_DIM + col]);
    }
    // 8 WMMAs per K-step: D = A x B + C
    #pragma unroll
    for (int m = 0; m < 4; ++m)
      #pragma unroll
      for (int n = 0; n < 2; ++n)
        acc[m][n] = __builtin_amdgcn_wmma_f32_16x16x4_f32(
            /*neg_a=*/false, a[m], /*neg_b=*/false, b[n],
            /*c_mod=*/(short)0, acc[m][n], /*reuse_a=*/false, /*reuse_b=*/false);
  }

  // Write back per C/D layout: vgpr r -> row (m*16 + r + 8*half), col (n-strip + l15)
  #pragma unroll
  for (int m = 0; m < 4; ++m) {
    #pragma unroll
    for (int n = 0; n < 2; ++n) {
      const int col = n0 + n * 16 + l15;
      #pragma unroll
      for (int r = 0; r < 8; ++r) {
        const int row = m * 16 + r + 8 * half;
        H[(size_t)row * N_DIM + col] = acc[m][n][r];
      }
    }
  }
}

// ---------------------------------------------------------------------------
// Phase 2: per spatial position p (1024 blocks), 64 threads (thread i = row i).
// Stage the 64x20 slice h[:, p, :] in LDS, compute
//   D[i][j] = exp(-sum_k |h_i[k]-h_j[k]|),
//   out[i]  = sum_{j>i} D[i][j] + sum_{t<i} D[t][t+1].
// ~84M VALU ops total: negligible next to phase 1's 671 MB stream.
// ---------------------------------------------------------------------------
__global__ __launch_bounds__(64) void pairwise_l1_exp(const float* __restrict__ H,
                                                      float* __restrict__ out) {
  __shared__ float hs[B_SZ][HID];
  __shared__ float arow[B_SZ];
  const int p = blockIdx.x;
  const int i = threadIdx.x;

  float hi[HID];
  #pragma unroll
  for (int k = 0; k < HID; ++k) {
    const float v = H[(size_t)i * N_DIM + p * HID + k];
    hi[k]    = v;
    hs[i][k] = v;
  }
  arow[i] = 0.0f;
  __syncthreads();

  float upper = 0.0f;
  for (int j = 0; j < B_SZ; ++j) {
    float l1 = 0.0f;
    #pragma unroll
    for (int k = 0; k < HID; ++k)
      l1 += fabsf(hi[k] - hs[j][k]);
    const float d = __expf(-l1);
    if (j > i)     upper += d;
    if (j == i + 1) arow[i] = d;   // superdiagonal D[i, i+1]
  }
  __syncthreads();

  float prefix = 0.0f;
  for (int t = 0; t < i; ++t) prefix += arow[t];

  out[(size_t)i * HW + p] = upper + prefix;
}

extern "C" void kernel_launch(void* const* d_in, const int* in_sizes, int n_in,
                              void* d_out, int out_size, void* d_ws, size_t ws_size,
                              hipStream_t stream) {
  (void)in_sizes; (void)n_in; (void)out_size; (void)ws_size;
  const float* x = (const float*)d_in[0];   // (64, 8, 32, 32) == (64, 8192) row-major
  const float* T = (const float*)d_in[1];   // (8192, 20480) row-major
  float*       o = (float*)d_out;           // (64, 1, 32, 32) == (64, 1024)
  float*       H = (float*)d_ws;            // (64, 20480) intermediate, 5.24 MB

  // 640 waves cover N=20480 exactly: 80 blocks x 8 waves, EXEC all-ones.
  gemm_wmma_f32<<<80, 256, 0, stream>>>(x, T, H);
  pairwise_l1_exp<<<HW, B_SZ, 0, stream>>>(H, o);
}